// Lift3DReplacementGraspNet_16733192585548
// MI455X (gfx1250) — compile-verified
//
#include <hip/hip_runtime.h>
#include <stdint.h>

typedef float v2f __attribute__((ext_vector_type(2)));
typedef float v8f __attribute__((ext_vector_type(8)));

constexpr int B_   = 16;
constexpr int S_   = 1024;
constexpr int N_   = 4096;
constexpr int C_   = 768;
constexpr int OUT_ = 256;

constexpr int KC   = 32;        // K chunk staged in LDS per pipeline stage
constexpr int NS   = 32;        // seed (column) tile per block
constexpr int WSTR = 36;        // W tile row stride (dwords): mult of 4 (b128 LDS align),
                                // gcd(36,64)=4 -> 16 lanes hit 16 distinct banks
constexpr int GSTR = NS + 1;    // gathered-feature tile row stride (dwords)
constexpr int WSIZE = OUT_ * WSTR;       // 9216 dwords
constexpr int GSIZE = KC * GSTR;         // 1056 dwords
constexpr int BUFSZ = WSIZE + GSIZE;     // 10272 dwords per pipeline buffer

// ---------------------------------------------------------------------------
// Kernel 1: nearest-neighbor argmin.  One block = (batch, 256 seeds).
// Full p_last[b] (4096 x 3 = 48KB) staged in LDS; every thread scans all N.
// ---------------------------------------------------------------------------
__global__ __launch_bounds__(256)
void nn_argmin_kernel(const float* __restrict__ seed,
                      const float* __restrict__ pl,
                      int* __restrict__ idx_out) {
    __shared__ float ps[N_ * 3];

    const int bb  = blockIdx.y;
    const int tid = threadIdx.x;

    const float* pg = pl + (size_t)bb * N_ * 3;
    for (int lin = tid; lin < N_ * 3; lin += 256) ps[lin] = pg[lin];
    __syncthreads();

    const int s = blockIdx.x * 256 + tid;
    const float* sp = seed + ((size_t)bb * S_ + s) * 3;
    const float sx = sp[0], sy = sp[1], sz = sp[2];

    float best = 3.4e38f;
    int   bi   = 0;
    for (int n = 0; n < N_; ++n) {
        const float dx = ps[n * 3 + 0] - sx;
        const float dy = ps[n * 3 + 1] - sy;
        const float dz = ps[n * 3 + 2] - sz;
        const float d2 = dx * dx + dy * dy + dz * dz;
        if (d2 < best) { best = d2; bi = n; }   // strict <  ==  first-occurrence argmin
    }
    idx_out[(size_t)bb * S_ + s] = bi;
}

// ---------------------------------------------------------------------------
// Async global->LDS copies (CDNA5, tracked by ASYNCcnt).  GVS addressing:
//   each lane copies from the global address saddr plus its vaddr to the LDS address in its vdst
// ---------------------------------------------------------------------------
__device__ __forceinline__ void async_copy_b32(uint32_t lds_addr, const void* sbase,
                                               uint32_t voff) {
    asm volatile("global_load_async_to_lds_b32 %0, %1, %2"
                 :: "v"(lds_addr), "v"(voff), "s"(sbase) : "memory");
}
__device__ __forceinline__ void async_copy_b128(uint32_t lds_addr, const void* sbase,
                                                uint32_t voff) {
    asm volatile("global_load_async_to_lds_b128 %0, %1, %2"
                 :: "v"(lds_addr), "v"(voff), "s"(sbase) : "memory");
}
__device__ __forceinline__ void wait_async0() {
    asm volatile("s_wait_asynccnt 0x0" ::: "memory");
}

// ---------------------------------------------------------------------------
// Kernel 2: gather + 1x1-conv GEMM via V_WMMA_F32_16X16X4_F32 with a
// double-buffered async-to-LDS pipeline.
// Block = 256 threads (8 wave32s) computes a 256(OUT) x 32(S) tile of one batch.
// Wave w: m-group = (w&3)*64 (4 stacked 16x16 tiles), n-group = (w>>2)*16.
// ---------------------------------------------------------------------------
__global__ __launch_bounds__(256)
void gather_gemm_wmma_kernel(const float* __restrict__ fg,    // [B,C,N]
                             const float* __restrict__ Wg,    // [OUT,C]
                             const float* __restrict__ bias,  // [OUT]
                             const int*   __restrict__ idx,   // [B,S]
                             float* __restrict__ out) {       // [B,OUT,S]
    __shared__ __attribute__((aligned(16))) float smem[2 * BUFSZ];  // ~80 KB

    const int bb   = blockIdx.y;
    const int s0   = blockIdx.x * NS;
    const int tid  = threadIdx.x;
    const int lane = tid & 31;
    const int wv   = tid >> 5;
    const int m_base = (wv & 3) * 64;
    const int n_base = (wv >> 2) * 16;

    // Per-thread gather index: staging assigns column n = tid&31 to this thread
    // for every chunk, so one index load covers all stages.
    const int nn = idx[(size_t)bb * S_ + s0 + (tid & 31)];

    const float* fgb = fg + (size_t)bb * C_ * N_;   // batch base for the gather
    const uint32_t smem_base = (uint32_t)(uintptr_t)smem;   // LDS byte offset

    // Stage one K-chunk into pipeline buffer `buf` (pure async issue, no wait).
    auto issue_stage = [&](int buf, int kc0) {
        const uint32_t wbase = smem_base + (uint32_t)(buf * BUFSZ) * 4u;
        const uint32_t gbase = wbase + (uint32_t)WSIZE * 4u;
        // W chunk [OUT][KC]: 8 x b128 per thread, 128B-coalesced per 8 lanes.
        const int kseg = (tid & 7) * 4;
        const int o0   = tid >> 3;
        #pragma unroll
        for (int i = 0; i < 8; ++i) {
            const int o = o0 + i * 32;
            async_copy_b128(wbase + (uint32_t)(o * WSTR + kseg) * 4u,
                            Wg, (uint32_t)(o * C_ + kc0 + kseg) * 4u);
        }
        // Gathered features [KC][NS]: scattered per-lane b32 copies.
        const int n  = tid & 31;
        const int k0 = tid >> 5;
        #pragma unroll
        for (int i = 0; i < 4; ++i) {
            const int k = k0 + i * 8;
            async_copy_b32(gbase + (uint32_t)(k * GSTR + n) * 4u,
                           fgb, (uint32_t)((kc0 + k) * N_ + nn) * 4u);
        }
    };

    v8f acc[4];
    const v8f vzero = {};
    acc[0] = vzero; acc[1] = vzero; acc[2] = vzero; acc[3] = vzero;

    constexpr int NCHUNK = C_ / KC;     // 24
    issue_stage(0, 0);

    int buf = 0;
    for (int ic = 0; ic < NCHUNK; ++ic) {
        wait_async0();          // my async writes for chunk `ic` have landed
        __syncthreads();        // everyone's have -> buffer `buf` is valid

        if (ic + 1 < NCHUNK) issue_stage(buf ^ 1, (ic + 1) * KC);

        const float* Wl = smem + buf * BUFSZ;
        const float* Gl = Wl + WSIZE;

        #pragma unroll
        for (int kk = 0; kk < KC / 4; ++kk) {
            // 16x4 f32 A layout: lanes 0-15 hold K=kk*4+{0,1}; lanes 16-31 K=kk*4+{2,3}
            const int krow = kk * 4 + ((lane >> 4) << 1);

            v2f bf;   // B is 4x16: row striped across lanes within a VGPR
            bf.x = Gl[(krow    ) * GSTR + n_base + (lane & 15)];
            bf.y = Gl[(krow + 1) * GSTR + n_base + (lane & 15)];

            #pragma unroll
            for (int t = 0; t < 4; ++t) {
                const int m = m_base + t * 16 + (lane & 15);
                v2f af;
                af.x = Wl[m * WSTR + krow];
                af.y = Wl[m * WSTR + krow + 1];
                acc[t] = __builtin_amdgcn_wmma_f32_16x16x4_f32(
                    /*neg_a=*/false, af, /*neg_b=*/false, bf,
                    /*c_mod=*/(short)0, acc[t],
                    /*reuse_a=*/false, /*reuse_b=*/false);
            }
        }
        // Per-wave LDS reads are dscnt-waited before their consuming WMMAs, so
        // passing next iteration's barrier guarantees this buffer is reusable.
        buf ^= 1;
    }

    // C/D 16x16 f32 layout: VGPR r -> M = r (lanes 0-15) or M = 8+r (lanes 16-31); N = lane&15
    const int mh  = (lane >> 4) * 8;
    const int col = s0 + n_base + (lane & 15);
    #pragma unroll
    for (int t = 0; t < 4; ++t) {
        #pragma unroll
        for (int r = 0; r < 8; ++r) {
            const int o = m_base + t * 16 + mh + r;
            out[((size_t)bb * OUT_ + o) * S_ + col] = acc[t][r] + bias[o];
        }
    }
}

// ---------------------------------------------------------------------------
extern "C" void kernel_launch(void* const* d_in, const int* in_sizes, int n_in,
                              void* d_out, int out_size, void* d_ws, size_t ws_size,
                              hipStream_t stream) {
    const float* seed = (const float*)d_in[0];   // [B,S,3]
    const float* pl   = (const float*)d_in[1];   // [B,N,3]
    const float* fg   = (const float*)d_in[2];   // [B,C,N]
    const float* Wg   = (const float*)d_in[3];   // [OUT,C]
    const float* bias = (const float*)d_in[4];   // [OUT]
    float* out = (float*)d_out;                  // [B,OUT,S]
    int*   idx = (int*)d_ws;                     // B*S ints = 64KB scratch

    dim3 g1(S_ / 256, B_);
    nn_argmin_kernel<<<g1, 256, 0, stream>>>(seed, pl, idx);

    dim3 g2(S_ / NS, B_);
    gather_gemm_wmma_kernel<<<g2, 256, 0, stream>>>(fg, Wg, bias, idx, out);
}